// YOLOv1_Loss_33492154974230
// MI455X (gfx1250) — compile-verified
//
#include <hip/hip_runtime.h>

typedef float v2f __attribute__((ext_vector_type(2)));
typedef float v8f __attribute__((ext_vector_type(8)));

#define K_GRID 64.0f
#define K_IMG  448.0f

__device__ __forceinline__ float yolo_iou(float b0, float b1, float b2, float b3,
                                          float gx, float gy,
                                          float g5, float g6, float g7, float g8) {
    float cx = truncf((gx + b0) * K_GRID);
    float cy = truncf((gy + b1) * K_GRID);
    float w  = truncf(b2 * K_IMG);
    float h  = truncf(b3 * K_IMG);
    float x0 = fmaxf(0.0f, cx - 0.5f * w);
    float y0 = fmaxf(0.0f, cy - 0.5f * h);
    float x1 = fminf(K_IMG - 1.0f, cx + 0.5f * w);
    float y1 = fminf(K_IMG - 1.0f, cy + 0.5f * h);
    float pa = (x1 - x0) * (y1 - y0);
    float ga = (g7 - g5) * (g8 - g6);
    float lx = fmaxf(x0, g5), rx = fminf(x1, g7);
    float uy = fmaxf(y0, g6), dy = fminf(y1, g8);
    float inter = ((rx < lx) || (dy < uy)) ? 0.0f : truncf((rx - lx) * (dy - uy));
    return inter / (pa + ga - inter);
}

// Stage 1: per-row loss terms + per-block 7-channel partial sums.
// Partials are written as 16-float records per block (channels 7..15 = 0.0)
// so stage 2 can consume them as 4x16 WMMA B-tiles.
__global__ __launch_bounds__(256)
void yolo_loss_partial(const float* __restrict__ pred,
                       const float* __restrict__ gt,
                       const unsigned char* __restrict__ mask,
                       float* __restrict__ partials,
                       int nrows) {
    float acc[7] = {0.f, 0.f, 0.f, 0.f, 0.f, 0.f, 0.f};

    const int stride = gridDim.x * blockDim.x;
    for (int r = blockIdx.x * blockDim.x + threadIdx.x; r < nrows; r += stride) {
        const float2* p2 = (const float2*)(pred + (size_t)r * 30);  // 8B-aligned rows
        const float4* g4 = (const float4*)(gt + (size_t)r * 32);    // 16B-aligned rows

        float pv[30], gv[32];
#pragma unroll
        for (int i = 0; i < 15; ++i) { float2 t = p2[i]; pv[2*i] = t.x; pv[2*i+1] = t.y; }
#pragma unroll
        for (int i = 0; i < 8; ++i) {
            float4 t = g4[i];
            gv[4*i] = t.x; gv[4*i+1] = t.y; gv[4*i+2] = t.z; gv[4*i+3] = t.w;
        }

        float m  = mask[r] ? 1.0f : 0.0f;
        float nm = 1.0f - m;

        float iou1 = yolo_iou(pv[0], pv[1], pv[2], pv[3],
                              gv[30], gv[31], gv[5], gv[6], gv[7], gv[8]);
        float iou2 = yolo_iou(pv[5], pv[6], pv[7], pv[8],
                              gv[30], gv[31], gv[5], gv[6], gv[7], gv[8]);
        bool  use2 = !(iou1 > iou2);                   // matches jnp.where semantics
        float best = (iou1 > iou2) ? iou1 : iou2;

        const float* bp = use2 ? (pv + 5) : (pv + 0);  // responsible box
        const float* bn = use2 ? (pv + 0) : (pv + 5);  // other box

        acc[0] += best * m;     // iou_sum
        acc[1] += m;            // object_num

        float cls = 0.0f;
#pragma unroll
        for (int c = 0; c < 20; ++c) {
            float d = gv[10 + c] - pv[10 + c];
            cls = fmaf(d, d, cls);
        }
        acc[2] += m * cls;      // class SSE

        float dx = bp[0] - gv[0];
        float dy = bp[1] - gv[1];
        acc[3] += m * (dx * dx + dy * dy);             // xy coord SSE

        const float eps = 1e-8f;
        float sw = sqrtf(bp[2] + eps) - sqrtf(gv[2] + eps);
        float sh = sqrtf(bp[3] + eps) - sqrtf(gv[3] + eps);
        acc[4] += m * (sw * sw + sh * sh);             // wh sqrt SSE

        float pc = bp[4] - 1.0f;
        acc[5] += m * pc * pc;                         // positive conf

        acc[6] += nm * (pv[4] * pv[4] + pv[9] * pv[9]) + m * bn[4] * bn[4]; // negative conf
    }

    // wave32 tree reduction per channel
#pragma unroll
    for (int c = 0; c < 7; ++c)
#pragma unroll
        for (int o = 16; o > 0; o >>= 1)
            acc[c] += __shfl_xor(acc[c], o, 32);

    __shared__ float red[8][8];   // 8 waves x (7 channels, padded)
    const int wave = threadIdx.x >> 5;
    const int lane = threadIdx.x & 31;
    if (lane == 0) {
#pragma unroll
        for (int c = 0; c < 7; ++c) red[wave][c] = acc[c];
    }
    __syncthreads();

    if (threadIdx.x < 16) {
        float v = 0.0f;
        if (threadIdx.x < 7) {
#pragma unroll
            for (int w = 0; w < 8; ++w) v += red[w][threadIdx.x];
        }
        partials[(size_t)blockIdx.x * 16 + threadIdx.x] = v;  // cols 7..15 -> 0.0
    }
}

// Stage 2: single wave32. Reduce nblocks x 16 partial records with
// V_WMMA_F32_16X16X4_F32: A = ones(16x4), B = 4x16 tile of partials,
// D[i][j] = sum_k B[k][j] + C[i][j]  ->  every row of C holds channel totals.
__global__ __launch_bounds__(32)
void yolo_loss_final(const float* __restrict__ partials,
                     float* __restrict__ out,
                     int ntiles, float inv_batch) {
    const int lane = threadIdx.x;           // 0..31, EXEC all ones (uniform flow)

    v8f c = {};                             // f32 accumulator, 8 VGPRs
    v2f a; a.x = 1.0f; a.y = 1.0f;          // A = ones(16x4)

    // B layout (32-bit 4x16): lanes 0-15 hold K=0 (b.x) / K=1 (b.y) at col=lane,
    // lanes 16-31 hold K=2 / K=3 at col=lane-16.
    const int boff = ((lane & 16) << 1) + (lane & 15);  // element within 64-float tile

    for (int t = 0; t < ntiles; ++t) {
        const float* tp = partials + (size_t)t * 64;
        v2f b;
        b.x = tp[boff];
        b.y = tp[boff + 16];
        c = __builtin_amdgcn_wmma_f32_16x16x4_f32(
                /*neg_a=*/false, a, /*neg_b=*/false, b,
                /*c_mod=*/(short)0, c, /*reuse_a=*/false, /*reuse_b=*/false);
    }

    // Row 0 (or row 8 for lanes 16..31 -- identical rows) lives in c[0].
    __shared__ float sums[16];
    sums[lane & 15] = c[0];
    __syncthreads();

    if (lane == 0) {
        float s_iou = sums[0], s_m  = sums[1], s_cls = sums[2];
        float s_xy  = sums[3], s_wh = sums[4], s_pc  = sums[5], s_nc = sums[6];
        float loss_class = s_cls * inv_batch;                       // POS_CLS = 1
        float loss_coord = 5.0f * (s_xy + s_wh) * inv_batch;        // L_COORD = 5
        float loss_pos   = s_pc * inv_batch;                        // POS_CONF = 1
        float loss_neg   = 0.5f * s_nc * inv_batch;                 // L_NOOBJ = 0.5
        out[0] = loss_class + loss_coord + loss_pos + loss_neg;
        out[1] = loss_coord;
        out[2] = loss_pos;
        out[3] = loss_neg;
        out[4] = loss_class;
        out[5] = s_iou;
        out[6] = s_m;
    }
}

extern "C" void kernel_launch(void* const* d_in, const int* in_sizes, int n_in,
                              void* d_out, int out_size, void* d_ws, size_t ws_size,
                              hipStream_t stream) {
    const float*         pred = (const float*)d_in[0];          // (B,7,7,30) f32
    const float*         gt   = (const float*)d_in[1];          // (B,7,7,32) f32
    const unsigned char* mask = (const unsigned char*)d_in[2];  // (B,7,7) bool
    float*               out  = (float*)d_out;                  // 7 f32
    float*               partials = (float*)d_ws;

    const int   nrows     = in_sizes[2];            // B*49
    const int   batch     = nrows / 49;
    const float inv_batch = 1.0f / (float)batch;

    int nblocks = 1024;                             // multiple of 4 (WMMA tiles)
    size_t need = (size_t)nblocks * 16 * sizeof(float);
    if (need > ws_size) {
        nblocks = (int)(ws_size / (16 * sizeof(float))) & ~3;
        if (nblocks < 4) nblocks = 4;
    }

    yolo_loss_partial<<<nblocks, 256, 0, stream>>>(pred, gt, mask, partials, nrows);
    yolo_loss_final<<<1, 32, 0, stream>>>(partials, out, nblocks / 4, inv_batch);
}